// NegSimHead_80865644249688
// MI455X (gfx1250) — compile-verified
//
#include <hip/hip_runtime.h>

typedef __attribute__((ext_vector_type(2))) float v2f;
typedef __attribute__((ext_vector_type(8))) float v8f;

#define NROWS 8192
#define DIMK  256

// GEMM tiling: block tile 128(M) x 64(N), 8 waves each owning a 32x32 macro-tile (2x2 wmma tiles)
#define BM        128
#define BN        64
#define BK        32
#define LDSTRIDE  36                      // 32 + 4 pad floats: 4-bank rotation/row, 16B aligned
#define A_TILE    (BM * LDSTRIDE)         // floats
#define B_TILE    (BN * LDSTRIDE)         // floats
#define BUF_FLOATS (A_TILE + B_TILE)      // 6912 floats = 27648 B per buffer

// ---------------------------------------------------------------- zero counts
__global__ void zero_counts_kernel(unsigned int* __restrict__ counts) {
    int i = blockIdx.x * blockDim.x + threadIdx.x;
    if (i < NROWS) counts[i] = 0u;
}

// ---------------------------------------------------------------- diag: V[i].T[i]
__global__ void diag_kernel(const float* __restrict__ V,
                            const float* __restrict__ T,
                            float* __restrict__ diag) {
    int row = blockIdx.x * blockDim.x + threadIdx.x;
    if (row >= NROWS) return;
    const float4* vp = (const float4*)(V + (size_t)row * DIMK);
    const float4* tp = (const float4*)(T + (size_t)row * DIMK);
    float acc = 0.0f;
#pragma unroll 8
    for (int q = 0; q < DIMK / 4; ++q) {
        float4 a = vp[q];
        float4 b = tp[q];
        acc += a.x * b.x + a.y * b.y + a.z * b.z + a.w * b.w;
    }
    diag[row] = acc;
}

// ---------------------------------------------------------------- fused GEMM + rank-count
__global__ __launch_bounds__(256) void sim_count_kernel(
    const float* __restrict__ V, const float* __restrict__ T,
    const float* __restrict__ diag, unsigned int* __restrict__ counts) {

    __shared__ float smem[2 * BUF_FLOATS];    // double-buffered A|B staging, 54 KB

    const int tid   = threadIdx.x;
    const int lane  = tid & 31;
    const int wave  = tid >> 5;               // 0..7
    const int waveM = wave & 3;               // 4 waves along M  -> 4*32 = 128 rows
    const int waveN = wave >> 2;              // 2 waves along N  -> 2*32 = 64 cols
    const int half  = lane >> 4;              // K-pair select per 32-bit 16x4 layout
    const int l16   = lane & 15;

    const int rowBase = blockIdx.y * BM;
    const int colBase = blockIdx.x * BN;

    // Raw LDS byte offset of smem (low 32 bits of the flat address are the LDS address).
    const unsigned int ldsAbs = (unsigned int)(uintptr_t)(void*)smem;

    // Per-thread async-copy geometry (chunk-invariant): rows of BK=32 floats = 8 float4.
    // A chunk: 128 rows x 8 float4 = 1024 xfers (4/thread); B chunk: 64 x 8 = 512 (2/thread).
    unsigned int aLds[4], aOff[4], bLds[2], bOff[2];
#pragma unroll
    for (int r = 0; r < 4; ++r) {
        int idx  = tid + r * 256;             // 0..1023
        int arow = idx >> 3;                  // 0..127
        int ac4  = idx & 7;                   // float4 column
        aLds[r] = (unsigned int)(arow * (LDSTRIDE * 4) + ac4 * 16);
        aOff[r] = (unsigned int)(arow * (DIMK * 4) + ac4 * 16);
    }
#pragma unroll
    for (int r = 0; r < 2; ++r) {
        int idx  = tid + r * 256;             // 0..511
        int brow = idx >> 3;                  // 0..63
        int bc4  = idx & 7;
        bLds[r] = (unsigned int)(brow * (LDSTRIDE * 4) + bc4 * 16);
        bOff[r] = (unsigned int)(brow * (DIMK * 4) + bc4 * 16);
    }

    // Stage K-chunk `kc` into buffer `p` via async DMA (6 ops/thread, ASYNCcnt-tracked).
    auto stage = [&](int p, int kc) {
        const float* aBase = V + (size_t)rowBase * DIMK + kc;   // uniform -> SGPR pair
        const float* bBase = T + (size_t)colBase * DIMK + kc;   // uniform -> SGPR pair
        const unsigned int abuf = ldsAbs + (unsigned int)(p * (BUF_FLOATS * 4));
        const unsigned int bbuf = abuf + (unsigned int)(A_TILE * 4);
#pragma unroll
        for (int r = 0; r < 4; ++r) {
            unsigned int l = abuf + aLds[r];
            asm volatile("global_load_async_to_lds_b128 %0, %1, %2"
                         :: "v"(l), "v"(aOff[r]), "s"(aBase) : "memory");
        }
#pragma unroll
        for (int r = 0; r < 2; ++r) {
            unsigned int l = bbuf + bLds[r];
            asm volatile("global_load_async_to_lds_b128 %0, %1, %2"
                         :: "v"(l), "v"(bOff[r]), "s"(bBase) : "memory");
        }
    };

    v8f acc00 = {}, acc01 = {}, acc10 = {}, acc11 = {};

    stage(0, 0);                              // prime buffer 0

    for (int kc = 0; kc < DIMK; kc += BK) {
        const int p = (kc / BK) & 1;
        if (kc + BK < DIMK) {
            stage(p ^ 1, kc + BK);            // prefetch next chunk into other buffer
            // wait for current chunk's 6 copies; next chunk's 6 may stay in flight
            asm volatile("s_wait_asynccnt 0x6" ::: "memory");
        } else {
            asm volatile("s_wait_asynccnt 0x0" ::: "memory");
        }
        __syncthreads();                      // all waves' DMA for buffer p visible

        const float* sA  = smem + p * BUF_FLOATS;
        const float* sB  = sA + A_TILE;
        const float* pa0 = &sA[(waveM * 32 + l16) * LDSTRIDE + 2 * half];
        const float* pa1 = pa0 + 16 * LDSTRIDE;
        const float* pb0 = &sB[(waveN * 32 + l16) * LDSTRIDE + 2 * half];
        const float* pb1 = pb0 + 16 * LDSTRIDE;

#pragma unroll
        for (int kk = 0; kk < BK; kk += 4) {
            v2f a0 = *(const v2f*)(pa0 + kk);
            v2f a1 = *(const v2f*)(pa1 + kk);
            v2f b0 = *(const v2f*)(pb0 + kk);
            v2f b1 = *(const v2f*)(pb1 + kk);
            // 4 independent accumulation chains; each fragment feeds 2 wmmas
            acc00 = __builtin_amdgcn_wmma_f32_16x16x4_f32(false, a0, false, b0, (short)0, acc00, false, false);
            acc01 = __builtin_amdgcn_wmma_f32_16x16x4_f32(false, a0, false, b1, (short)0, acc01, false, false);
            acc10 = __builtin_amdgcn_wmma_f32_16x16x4_f32(false, a1, false, b0, (short)0, acc10, false, false);
            acc11 = __builtin_amdgcn_wmma_f32_16x16x4_f32(false, a1, false, b1, (short)0, acc11, false, false);
        }
        __syncthreads();                      // buffer p free for reuse
    }

    // C/D layout: VGPR r -> lanes 0-15: (M=r, N=lane), lanes 16-31: (M=r+8, N=lane-16)
    const int mBase = rowBase + waveM * 32;
#pragma unroll
    for (int i = 0; i < 2; ++i) {             // two M sub-tiles (rows +0 / +16)
        v8f aT = (i == 0) ? acc00 : acc10;    // N sub-tile 0
        v8f bT = (i == 0) ? acc01 : acc11;    // N sub-tile 1
        const int mb = mBase + i * 16;
#pragma unroll
        for (int r = 0; r < 8; ++r) {
            const int rowLo = mb + r;
            const int rowHi = mb + 8 + r;
            const float dref = diag[half ? rowHi : rowLo];
            unsigned int m0 = __builtin_amdgcn_ballot_w32(aT[r] > dref);
            unsigned int m1 = __builtin_amdgcn_ballot_w32(bT[r] > dref);
            if (lane == 0) {
                atomicAdd(&counts[rowLo],
                          (unsigned int)(__popc(m0 & 0xFFFFu) + __popc(m1 & 0xFFFFu)));
                atomicAdd(&counts[rowHi],
                          (unsigned int)(__popc(m0 >> 16) + __popc(m1 >> 16)));
            }
        }
    }
}

// ---------------------------------------------------------------- final scalars
__global__ void reduce_kernel(const float* __restrict__ diag,
                              const unsigned int* __restrict__ counts,
                              float* __restrict__ out) {
    __shared__ float s0[256], s1[256], s2[256], s3[256], s4[256];
    const int tid = threadIdx.x;
    float dsum = 0.f, c1 = 0.f, c5 = 0.f, c10 = 0.f, psum = 0.f;
    for (int i = tid; i < NROWS; i += 256) {
        dsum += diag[i];
        unsigned int p = counts[i];
        c1   += (p < 1u)  ? 1.f : 0.f;
        c5   += (p < 5u)  ? 1.f : 0.f;
        c10  += (p < 10u) ? 1.f : 0.f;
        psum += (float)p;
    }
    s0[tid] = dsum; s1[tid] = c1; s2[tid] = c5; s3[tid] = c10; s4[tid] = psum;
    __syncthreads();
    for (int s = 128; s > 0; s >>= 1) {
        if (tid < s) {
            s0[tid] += s0[tid + s]; s1[tid] += s1[tid + s]; s2[tid] += s2[tid + s];
            s3[tid] += s3[tid + s]; s4[tid] += s4[tid + s];
        }
        __syncthreads();
    }
    if (tid == 0) {
        const float invN = 1.0f / (float)NROWS;
        out[0] = -s0[0] * invN;   // neg_sim_loss
        out[1] =  s1[0] * invN;   // recall@1
        out[2] =  s2[0] * invN;   // recall@5
        out[3] =  s3[0] * invN;   // recall@10
        out[4] =  s4[0] * invN;   // mean rank
    }
}

extern "C" void kernel_launch(void* const* d_in, const int* in_sizes, int n_in,
                              void* d_out, int out_size, void* d_ws, size_t ws_size,
                              hipStream_t stream) {
    const float* V = (const float*)d_in[0];   // v_feat [8192,256] f32
    const float* T = (const float*)d_in[1];   // t_feat [8192,256] f32
    float* out = (float*)d_out;               // 5 f32 scalars

    // workspace: diag[8192] f32, counts[8192] u32
    float*        diag   = (float*)d_ws;
    unsigned int* counts = (unsigned int*)((char*)d_ws + NROWS * sizeof(float));

    zero_counts_kernel<<<NROWS / 256, 256, 0, stream>>>(counts);
    diag_kernel<<<NROWS / 256, 256, 0, stream>>>(V, T, diag);

    dim3 grid(NROWS / BN, NROWS / BM);        // 128 x 64 blocks, 8 waves each
    sim_count_kernel<<<grid, 256, 0, stream>>>(V, T, diag, counts);

    reduce_kernel<<<1, 256, 0, stream>>>(diag, counts, out);
}